// EnhancedAdaptiveGate_75076028334313
// MI455X (gfx1250) — compile-verified
//
#include <hip/hip_runtime.h>
#include <math.h>

typedef float v2f __attribute__((ext_vector_type(2)));
typedef float v8f __attribute__((ext_vector_type(8)));

#define B_   64
#define L_   4096
#define C_   128
#define NF_  768      // 6*C
#define H1_  128
#define H2_  32
#define NCHUNK 32
#define CHUNK_L 128
#define RPT 32                 // rows per thread (CHUNK_L / 4 row-subgroups)
#define RECENT_START 3072      // max(1, L - L//4)
#define RECENT_N 1024
#define EPSF 1e-8f

// ---------------------------------------------------------------------------
// Kernel 1: streaming partial reduction over L. Bandwidth-bound: float4 loads,
// contiguous lanes -> 512B per wave load. Each thread owns 4 channels and 32
// consecutive rows so diffs are local (1 boundary row re-read per thread).
// ---------------------------------------------------------------------------
__global__ __launch_bounds__(128)
void stats_partial_kernel(const float* __restrict__ x, float* __restrict__ part) {
  const int chunk = blockIdx.x;          // 0..31
  const int b     = blockIdx.y;          // 0..63
  const int tid   = threadIdx.x;         // 0..127
  const int cg    = tid & 31;            // channel group: cols 4*cg..4*cg+3
  const int r     = tid >> 5;            // row subgroup 0..3
  const int l0    = chunk * CHUNK_L + r * RPT;
  const float* xp = x + ((size_t)b * L_ + l0) * C_ + cg * 4;

  float s1[4]={0,0,0,0}, s2[4]={0,0,0,0}, s3[4]={0,0,0,0};
  float sd2[4]={0,0,0,0}, sr[4]={0,0,0,0}, sr2[4]={0,0,0,0};
  float cur[4];

  {
    float4 v = *(const float4*)xp;
    cur[0]=v.x; cur[1]=v.y; cur[2]=v.z; cur[3]=v.w;
    const bool rec = (l0 >= RECENT_START);
    #pragma unroll
    for (int j=0;j<4;++j) {
      float vv=cur[j], v2=vv*vv;
      s1[j]+=vv; s2[j]+=v2; s3[j]+=v2*vv;
      if (rec){ sr[j]+=vv; sr2[j]+=v2; }
    }
  }
  #pragma unroll 4
  for (int i=1;i<RPT;++i) {
    float4 v = *(const float4*)(xp + (size_t)i * C_);
    float nv[4]={v.x,v.y,v.z,v.w};
    const bool rec = (l0 + i >= RECENT_START);
    #pragma unroll
    for (int j=0;j<4;++j) {
      float d = nv[j]-cur[j]; sd2[j] += d*d;
      float vv=nv[j], v2=vv*vv;
      s1[j]+=vv; s2[j]+=v2; s3[j]+=v2*vv;
      if (rec){ sr[j]+=vv; sr2[j]+=v2; }
      cur[j]=nv[j];
    }
  }
  if (l0 + RPT < L_) {  // boundary diff into next subgroup's first row
    float4 v = *(const float4*)(xp + (size_t)RPT * C_);
    float nv[4]={v.x,v.y,v.z,v.w};
    #pragma unroll
    for (int j=0;j<4;++j){ float d=nv[j]-cur[j]; sd2[j]+=d*d; }
  }

  __shared__ float red[4][6][C_];
  #pragma unroll
  for (int j=0;j<4;++j) {
    const int c = cg*4+j;
    red[r][0][c]=s1[j];  red[r][1][c]=s2[j];  red[r][2][c]=s3[j];
    red[r][3][c]=sd2[j]; red[r][4][c]=sr[j];  red[r][5][c]=sr2[j];
  }
  __syncthreads();
  const int c = tid;   // 0..127, one channel each
  #pragma unroll
  for (int st=0; st<6; ++st) {
    float t = red[0][st][c]+red[1][st][c]+red[2][st][c]+red[3][st][c];
    part[(((size_t)st*NCHUNK + chunk)*B_ + b)*C_ + c] = t;  // coalesced
  }
}

// ---------------------------------------------------------------------------
// Kernel 2: reduce 32 chunk-partials per (b,c); raw moments -> reference stats.
// ---------------------------------------------------------------------------
__global__ __launch_bounds__(128)
void stats_finalize_kernel(const float* __restrict__ x,
                           const float* __restrict__ part,
                           float* __restrict__ stats) {
  const int b = blockIdx.x;
  const int c = threadIdx.x;
  float t[6] = {0,0,0,0,0,0};
  for (int ch=0; ch<NCHUNK; ++ch) {
    #pragma unroll
    for (int st=0; st<6; ++st)
      t[st] += part[(((size_t)st*NCHUNK + ch)*B_ + b)*C_ + c];
  }
  const float n  = (float)L_;
  const float mean = t[0] / n;
  float var = (t[1] - n*mean*mean) / (n - 1.0f);
  var = fmaxf(var, 0.0f);
  const float stdv = sqrtf(var) + EPSF;                      // std = ddof1 + EPS
  const float m3 = t[2]/n - 3.0f*mean*(t[1]/n) + 2.0f*mean*mean*mean;
  const float skew = m3 / (stdv*stdv*stdv);

  // diff stats: sum(d) telescopes
  const float x0 = x[(size_t)b * L_ * C_ + c];
  const float xl = x[((size_t)b * L_ + (L_-1)) * C_ + c];
  const float nd = (float)(L_ - 1);
  const float md = (xl - x0) / nd;
  float vard = (t[3] - nd*md*md) / (nd - 1.0f);
  vard = fmaxf(vard, 0.0f);
  const float dstd = sqrtf(vard);                            // no EPS in reference

  const float nr = (float)RECENT_N;
  const float rmean = t[4] / nr;
  float rvar = (t[5] - nr*rmean*rmean) / (nr - 1.0f);
  rvar = fmaxf(rvar, 0.0f);
  const float rstd = sqrtf(rvar) + EPSF;

  float* sp = stats + (size_t)b * NF_;
  sp[0*C_ + c] = mean;
  sp[1*C_ + c] = stdv;
  sp[2*C_ + c] = skew;
  sp[3*C_ + c] = dstd;
  sp[4*C_ + c] = rmean;
  sp[5*C_ + c] = rstd;
}

// ---------------------------------------------------------------------------
// Kernel 3: fused 3-layer MLP with V_WMMA_F32_16X16X4_F32.
// One block, 8 wave32s. Fragment layouts per CDNA5 ISA 7.12.2:
//   A 16x4 f32: lane m = lane&15; vgpr pair = K {0,1} (lanes 0-15) / {2,3} (16-31)
//   B/C/D: row striped across lanes; C/D vgpr r = row (tm + r + 8*(lane>=16))
// ---------------------------------------------------------------------------
__device__ __forceinline__ float gelu_exact(float v) {
  return 0.5f * v * (1.0f + erff(v * 0.70710678118654752f));
}

__global__ __launch_bounds__(256)
void mlp_kernel(const float* __restrict__ stats,
                const float* __restrict__ W1, const float* __restrict__ b1,
                const float* __restrict__ W2, const float* __restrict__ b2,
                const float* __restrict__ W3, const float* __restrict__ b3,
                float* __restrict__ out) {
  __shared__ float h1[B_ * H1_];   // 32 KB
  __shared__ float h2[B_ * H2_];   //  8 KB
  const int tid  = threadIdx.x;
  const int wave = tid >> 5;       // 0..7
  const int lane = tid & 31;
  const int lr   = lane & 15;
  const int khi  = (lane >> 4) & 1;   // K offset 0 or 2 within quad
  const int moff = khi * 8;           // C/D row offset for lanes 16-31

  // ---- Layer 1: stats(64x768) @ W1(768x128) + b1, GELU -> h1 ----
  for (int t = wave; t < (B_/16)*(H1_/16); t += 8) {
    const int tm = (t >> 3) << 4;
    const int tn = (t & 7) << 4;
    v8f acc = {0.f,0.f,0.f,0.f,0.f,0.f,0.f,0.f};
    for (int k = 0; k < NF_; k += 4) {
      const int ka = k + khi*2;
      v2f a, bb;
      const float* ap = stats + (size_t)(tm + lr) * NF_ + ka;
      a[0] = ap[0]; a[1] = ap[1];
      const float* bp = W1 + (size_t)ka * H1_ + tn + lr;
      bb[0] = bp[0]; bb[1] = bp[H1_];
      acc = __builtin_amdgcn_wmma_f32_16x16x4_f32(false, a, false, bb,
                                                  (short)0, acc, false, false);
    }
    const int n = tn + lr;
    const float bias = b1[n];
    #pragma unroll
    for (int rr = 0; rr < 8; ++rr)
      h1[(tm + moff + rr) * H1_ + n] = gelu_exact(acc[rr] + bias);
  }
  __syncthreads();

  // ---- Layer 2: h1(64x128) @ W2(128x32) + b2, GELU -> h2 (8 tiles, 1/wave) ----
  {
    const int tm = (wave >> 1) << 4;
    const int tn = (wave & 1) << 4;
    v8f acc = {0.f,0.f,0.f,0.f,0.f,0.f,0.f,0.f};
    for (int k = 0; k < H1_; k += 4) {
      const int ka = k + khi*2;
      v2f a, bb;
      const float* ap = &h1[(tm + lr) * H1_ + ka];
      a[0] = ap[0]; a[1] = ap[1];
      const float* bp = W2 + (size_t)ka * H2_ + tn + lr;
      bb[0] = bp[0]; bb[1] = bp[H2_];
      acc = __builtin_amdgcn_wmma_f32_16x16x4_f32(false, a, false, bb,
                                                  (short)0, acc, false, false);
    }
    const int n = tn + lr;
    const float bias = b2[n];
    #pragma unroll
    for (int rr = 0; rr < 8; ++rr)
      h2[(tm + moff + rr) * H2_ + n] = gelu_exact(acc[rr] + bias);
  }
  __syncthreads();

  // ---- Layer 3: h2(64x32) @ W3(32x128) + b3, sigmoid -> out ----
  for (int t = wave; t < (B_/16)*(C_/16); t += 8) {
    const int tm = (t >> 3) << 4;
    const int tn = (t & 7) << 4;
    v8f acc = {0.f,0.f,0.f,0.f,0.f,0.f,0.f,0.f};
    for (int k = 0; k < H2_; k += 4) {
      const int ka = k + khi*2;
      v2f a, bb;
      const float* ap = &h2[(tm + lr) * H2_ + ka];
      a[0] = ap[0]; a[1] = ap[1];
      const float* bp = W3 + (size_t)ka * C_ + tn + lr;
      bb[0] = bp[0]; bb[1] = bp[C_];
      acc = __builtin_amdgcn_wmma_f32_16x16x4_f32(false, a, false, bb,
                                                  (short)0, acc, false, false);
    }
    const int n = tn + lr;
    const float bias = b3[n];
    #pragma unroll
    for (int rr = 0; rr < 8; ++rr) {
      const float v = acc[rr] + bias;
      out[(size_t)(tm + moff + rr) * C_ + n] = 1.0f / (1.0f + expf(-v));
    }
  }
}

// ---------------------------------------------------------------------------
extern "C" void kernel_launch(void* const* d_in, const int* in_sizes, int n_in,
                              void* d_out, int out_size, void* d_ws, size_t ws_size,
                              hipStream_t stream) {
  const float* x  = (const float*)d_in[0];
  const float* W1 = (const float*)d_in[1];
  const float* b1 = (const float*)d_in[2];
  const float* W2 = (const float*)d_in[3];
  const float* b2 = (const float*)d_in[4];
  const float* W3 = (const float*)d_in[5];
  const float* b3 = (const float*)d_in[6];
  float* out = (float*)d_out;

  float* part  = (float*)d_ws;                              // 6*32*64*128 floats
  float* stats = part + (size_t)6 * NCHUNK * B_ * C_;       // 64*768 floats

  stats_partial_kernel <<<dim3(NCHUNK, B_), 128, 0, stream>>>(x, part);
  stats_finalize_kernel<<<B_, C_, 0, stream>>>(x, part, stats);
  mlp_kernel<<<1, 256, 0, stream>>>(stats, W1, b1, W2, b2, W3, b3, out);
}